// mixMseLoss_64922725646764
// MI455X (gfx1250) — compile-verified
//
#include <hip/hip_runtime.h>

// CDNA5 / gfx1250 implementation of the greedy matched-MSE loss.
//
// Shape of the problem: B=2048 independent batches; each needs a 256x256
// f32 distance matrix and a strictly sequential 256-step greedy argmin scan.
// -> one workgroup (256 threads = 8 wave32) per batch.
// -> distance panels (16 rows x 256 cols) built with V_WMMA_F32_16X16X4_F32
//    using the augmented K=4 formulation:
//      A row j = ( tx, ty, |t|^2, 1 ),  B col k = ( -2px, -2py, 1, |p|^2 )
//      => (A x B)[j][k] = ||t_j - p_k||^2   exactly, in one WMMA, f32.
// -> wave 0 runs the greedy scan with shuffle-based min/argmin (first-min-wins
//    tie break on the index), used-mask kept in registers (8 bits per lane).
// -> per-batch partial sums -> deterministic tree reduction kernel.

typedef __attribute__((ext_vector_type(2))) float v2f;
typedef __attribute__((ext_vector_type(8))) float v8f;

#define NPTS 256
#define BIGV 66049.0f            // 257^2 sentinel from the reference
#define PANEL_STRIDE 258         // 258 % 8 == 2 -> M and M+8 half-waves hit
                                 // disjoint LDS bank windows (64 banks)

__global__ __launch_bounds__(256) void greedy_match_kernel(
    const float* __restrict__ inp,   // [B, 512] interleaved (x,y) input points
    const float* __restrict__ tgt,   // [B, 512] interleaved (x,y) target points
    float* __restrict__ partial)     // [B] per-batch SE sums
{
    __shared__ float sT[2 * NPTS];               // targets, interleaved
    __shared__ float sP[2 * NPTS];               // inputs, interleaved
    __shared__ float sPanel[16 * PANEL_STRIDE];  // 16 rows of D, padded

    const int tid   = threadIdx.x;
    const int b     = blockIdx.x;
    const int wave  = tid >> 5;
    const int lane  = tid & 31;
    const int row16 = lane & 15;   // M (or N) index within a 16-wide tile
    const int hi    = lane >> 4;   // 0: K=0/1 half, 1: K=2/3 half (A/B layout)

    // Stage both point sets into LDS (2 KB each).
    const float* tb = tgt + (size_t)b * (2 * NPTS);
    const float* pb = inp + (size_t)b * (2 * NPTS);
    sT[tid]        = tb[tid];
    sT[tid + 256]  = tb[tid + 256];
    sP[tid]        = pb[tid];
    sP[tid + 256]  = pb[tid + 256];
    __syncthreads();

    float    acc      = 0.0f;  // wave-0 SE accumulator
    unsigned usedMask = 0u;    // wave-0: bit i <-> input point k = i*32 + lane

    for (int jt = 0; jt < 16; ++jt) {
        // ---- A operand: target rows M = jt*16 + row16, augmented K=4 ----
        // VGPR0: lanes0-15 = K0 (tx),  lanes16-31 = K2 (|t|^2)
        // VGPR1: lanes0-15 = K1 (ty),  lanes16-31 = K3 (1.0)
        v2f a;
        {
            int   m  = jt * 16 + row16;
            float tx = sT[2 * m], ty = sT[2 * m + 1];
            if (hi == 0) { a[0] = tx;               a[1] = ty;   }
            else         { a[0] = tx * tx + ty * ty; a[1] = 1.0f; }
        }

        // ---- 16 N-tiles per panel, 2 per wave ----
        #pragma unroll
        for (int q = 0; q < 2; ++q) {
            int   nt = wave * 2 + q;
            int   n  = nt * 16 + row16;
            float px = sP[2 * n], py = sP[2 * n + 1];
            // B operand: VGPR0: K0 row (-2px) / K2 row (1.0)
            //            VGPR1: K1 row (-2py) / K3 row (|p|^2)
            v2f bb;
            if (hi == 0) { bb[0] = -2.0f * px; bb[1] = -2.0f * py;     }
            else         { bb[0] = 1.0f;       bb[1] = px * px + py * py; }

            v8f c = {};
            c = __builtin_amdgcn_wmma_f32_16x16x4_f32(
                    /*neg_a=*/false, a, /*neg_b=*/false, bb,
                    /*c_mod=*/(short)0, c, /*reuse_a=*/false, /*reuse_b=*/false);

            // C/D layout: VGPR r -> row (hi*8 + r), col = row16 within tile.
            int baseRow = hi * 8;
            int col     = nt * 16 + row16;
            #pragma unroll
            for (int r = 0; r < 8; ++r)
                sPanel[(baseRow + r) * PANEL_STRIDE + col] = c[r];
        }
        __syncthreads();

        // ---- wave 0: 16 greedy steps over this panel ----
        if (wave == 0) {
            for (int jj = 0; jj < 16; ++jj) {
                const float* rowD = &sPanel[jj * PANEL_STRIDE];
                float bestV = INFINITY;
                int   bestK = 0;
                // lane-local first-wins min over k = i*32 + lane (ascending k)
                #pragma unroll
                for (int i = 0; i < 8; ++i) {
                    int   k = i * 32 + lane;
                    float v = ((usedMask >> i) & 1u) ? INFINITY : rowD[k];
                    if (v < bestV) { bestV = v; bestK = k; }
                }
                // cross-lane min/argmin, tie-break to smaller k (= first min)
                #pragma unroll
                for (int off = 16; off >= 1; off >>= 1) {
                    float ov = __shfl_xor(bestV, off, 32);
                    int   ok = __shfl_xor(bestK, off, 32);
                    if (ov < bestV || (ov == bestV && ok < bestK)) {
                        bestV = ov; bestK = ok;
                    }
                }
                bool  hit  = bestV < BIGV;
                float se   = hit ? bestV : BIGV;   // min(m, 257^2)
                int   kSel = hit ? bestK : 0;      // k stays 0 if nothing beats BIG
                if ((kSel & 31) == lane)
                    usedMask |= (1u << (kSel >> 5));
                acc += se;
            }
        }
        __syncthreads();
    }

    if (tid == 0) partial[b] = acc;
}

__global__ __launch_bounds__(256) void reduce_loss_kernel(
    const float* __restrict__ partial, float* __restrict__ out, int B)
{
    __shared__ float s[256];
    float x = 0.0f;
    for (int i = threadIdx.x; i < B; i += 256) x += partial[i];
    s[threadIdx.x] = x;
    __syncthreads();
    #pragma unroll
    for (int o = 128; o >= 1; o >>= 1) {
        if ((int)threadIdx.x < o) s[threadIdx.x] += s[threadIdx.x + o];
        __syncthreads();
    }
    if (threadIdx.x == 0) out[0] = s[0] / (float)B / (float)(2 * NPTS);
}

extern "C" void kernel_launch(void* const* d_in, const int* in_sizes, int n_in,
                              void* d_out, int out_size, void* d_ws, size_t ws_size,
                              hipStream_t stream) {
    const float* inp = (const float*)d_in[0];   // "input"  [B,1,512] f32
    const float* tgt = (const float*)d_in[1];   // "targets" [B,1,512] f32
    float* partial   = (float*)d_ws;            // B floats of scratch
    float* out       = (float*)d_out;           // 1 float

    const int B = in_sizes[0] / (2 * NPTS);     // 2048 for the reference setup

    greedy_match_kernel<<<B, 256, 0, stream>>>(inp, tgt, partial);
    reduce_loss_kernel<<<1, 256, 0, stream>>>(partial, out, B);
}